// GraphConv_5162550690524
// MI455X (gfx1250) — compile-verified
//
#include <hip/hip_runtime.h>
#include <hip/hip_bf16.h>

// GCN 2-layer forward for MI455X (gfx1250, wave32).
//   deg[n]  = 1 + sum_{e: dst==n} w[e];  dinv = rsqrt(deg)
//   h1      = x @ W1                      (WMMA f32 16x16x4)
//   agg1[n] = dinv[n]^2*h1[n] + b1 + sum_e dinv[src]*w*dinv[dst]*h1[src]  (L2 atomics)
//   a1      = relu(agg1)   (leaky_relu(relu(x)) == relu(x))
//   h2      = a1 @ W2                     (WMMA f32 16x16x4)
//   out[n]  = dinv[n]^2*h2[n] + b2 + sum_e dinv[src]*w*dinv[dst]*h2[src]
//
// Bandwidth-bound in the edge scatter; all hot arrays (<70MB) are L2-resident
// (192MB L2), so f32 atomics resolve at L2 rate. GEMMs are tiny (2 GFLOP
// total) -> keep full fp32 precision via V_WMMA_F32_16X16X4_F32.

typedef float v2f __attribute__((ext_vector_type(2)));
typedef float v8f __attribute__((ext_vector_type(8)));

// ---------------- degree / normalization ----------------

__global__ void k_fill1(float* __restrict__ p, int n) {
  int i = blockIdx.x * blockDim.x + threadIdx.x;
  if (i < n) p[i] = 1.0f;  // self-loop weight
}

__global__ void k_deg_accum(const int* __restrict__ dst, const float* __restrict__ w,
                            float* __restrict__ deg, int E) {
  int i = blockIdx.x * blockDim.x + threadIdx.x;
  if (i < E) atomicAdd(&deg[dst[i]], w[i]);
}

__global__ void k_dinv(float* __restrict__ p, int n) {
  int i = blockIdx.x * blockDim.x + threadIdx.x;
  if (i < n) {
    float v = p[i];
    p[i] = v > 0.0f ? rsqrtf(v) : 0.0f;
  }
}

// ---------------- dense GEMM via f32 WMMA ----------------
// D[M x NCOLS] = A[M x K] @ B[K x NCOLS], all row-major f32.
// One wave computes one 16x16 tile of D. M must be a multiple of 16
// (100000 = 6250*16). All branches are wave-uniform -> EXEC all-1s for WMMA.
template <int K, int NCOLS>
__global__ void k_gemm_wmma(const float* __restrict__ A, const float* __restrict__ B,
                            float* __restrict__ D, int M) {
  const int lane = threadIdx.x & 31;
  const int wave = threadIdx.x >> 5;
  const int tilesN = NCOLS / 16;
  const int tilesM = M / 16;
  const int tile = blockIdx.x * (blockDim.x >> 5) + wave;  // wave-uniform
  if (tile >= tilesM * tilesN) return;                      // whole-wave exit

  const int tm = tile / tilesN;
  const int tn = tile - tm * tilesN;
  const int m0 = tm * 16;
  const int n0 = tn * 16;

  const int r    = lane & 15;  // row (A) / col (B,D) within tile
  const int half = lane >> 4;  // K sub-phase: half=0 -> K=0,1 ; half=1 -> K=2,3

  const float* __restrict__ arow = A + (size_t)(m0 + r) * K;
  const float* __restrict__ bcol = B + n0 + r;

  v8f c = {0.f, 0.f, 0.f, 0.f, 0.f, 0.f, 0.f, 0.f};

#pragma unroll
  for (int k0 = 0; k0 < K; k0 += 4) {
    const int kk = k0 + half * 2;
    // A 16x4: lane(half,r) holds A[m0+r][kk], A[m0+r][kk+1]  (8B-aligned)
    const float2 av = *(const float2*)(arow + kk);
    v2f a;  a.x = av.x;            a.y = av.y;
    // B 4x16: lane(half,r) holds B[kk][n0+r], B[kk+1][n0+r]
    v2f b;  b.x = bcol[(size_t)kk * NCOLS];
            b.y = bcol[(size_t)(kk + 1) * NCOLS];
    c = __builtin_amdgcn_wmma_f32_16x16x4_f32(
        /*neg_a=*/false, a, /*neg_b=*/false, b,
        /*c_mod=*/(short)0, c, /*reuse_a=*/false, /*reuse_b=*/false);
  }

  // C/D layout: VGPR i -> row m0 + 8*half + i, col n0 + r
  const int rowbase = m0 + half * 8;
#pragma unroll
  for (int i = 0; i < 8; ++i)
    D[(size_t)(rowbase + i) * NCOLS + n0 + r] = c[i];
}

// ---------------- aggregation ----------------

// agg[n][c] = dinv[n]^2 * h[n][c] + bias[c]   (self-loop + bias)
template <int C>
__global__ void k_agg_init(const float* __restrict__ h, const float* __restrict__ dinv,
                           const float* __restrict__ bias, float* __restrict__ agg, int n) {
  int i = blockIdx.x * blockDim.x + threadIdx.x;
  if (i < n * C) {
    const int node = i / C;
    const int ch   = i - node * C;
    const float s  = dinv[node];
    agg[i] = s * s * h[i] + bias[ch];
  }
}

// Per-edge scatter: each thread handles 4 channels of one edge.
// Consecutive lanes -> consecutive channels -> coalesced gather of h[src].
template <int C>
__global__ void k_edge_scatter(const int* __restrict__ src, const int* __restrict__ dst,
                               const float* __restrict__ w, const float* __restrict__ dinv,
                               const float* __restrict__ h, float* __restrict__ agg, int E) {
  constexpr int GROUPS = C / 4;
  const long long t = (long long)blockIdx.x * blockDim.x + threadIdx.x;
  const long long total = (long long)E * GROUPS;
  if (t >= total) return;
  const int e = (int)(t / GROUPS);   // GROUPS is a power of two -> shift
  const int g = (int)(t - (long long)e * GROUPS);
  const int s = src[e];
  const int d = dst[e];
  const float nm = dinv[s] * w[e] * dinv[d];
  const float4 hv = *(const float4*)(h + (size_t)s * C + g * 4);
  float* o = agg + (size_t)d * C + g * 4;
  atomicAdd(o + 0, nm * hv.x);
  atomicAdd(o + 1, nm * hv.y);
  atomicAdd(o + 2, nm * hv.z);
  atomicAdd(o + 3, nm * hv.w);
}

__global__ void k_relu(float* __restrict__ p, int n) {
  int i = blockIdx.x * blockDim.x + threadIdx.x;
  if (i < n) p[i] = fmaxf(p[i], 0.0f);
}

// ---------------- host launcher ----------------

static inline int cdiv(long long a, long long b) { return (int)((a + b - 1) / b); }

extern "C" void kernel_launch(void* const* d_in, const int* in_sizes, int n_in,
                              void* d_out, int out_size, void* d_ws, size_t ws_size,
                              hipStream_t stream) {
  constexpr int IN_CH = 128, HID_CH = 64, OUT_CH = 32;

  const float* x   = (const float*)d_in[0];
  const int*   ei  = (const int*)d_in[1];
  const float* ew  = (const float*)d_in[2];
  const float* W1  = (const float*)d_in[3];
  const float* b1  = (const float*)d_in[4];
  const float* W2  = (const float*)d_in[5];
  const float* b2  = (const float*)d_in[6];
  float*       out = (float*)d_out;

  const int N = in_sizes[0] / IN_CH;  // 100000 (multiple of 16)
  const int E = in_sizes[2];          // 1600000
  const int* src = ei;                // edge_index[0]
  const int* dst = ei + E;            // edge_index[1]

  // Workspace layout (bytes): dinv @0 (0.4MB), h @512KB (25.6MB, reused for
  // h1 then h2), agg1 after h (25.6MB). Total ~49.3MB.
  char*  ws    = (char*)d_ws;
  float* dinv  = (float*)ws;
  float* hbuf  = (float*)(ws + (size_t)(512 * 1024));
  float* agg1  = (float*)(ws + (size_t)(512 * 1024) + (size_t)N * HID_CH * sizeof(float));

  const int BLK = 256;

  // 1) degree with self-loops -> dinv
  k_fill1<<<cdiv(N, BLK), BLK, 0, stream>>>(dinv, N);
  k_deg_accum<<<cdiv(E, BLK), BLK, 0, stream>>>(dst, ew, dinv, E);
  k_dinv<<<cdiv(N, BLK), BLK, 0, stream>>>(dinv, N);

  // 2) h1 = x @ W1   (WMMA: 6250 * 4 tiles, 8 waves/block)
  {
    const int waves = (N / 16) * (HID_CH / 16);
    k_gemm_wmma<IN_CH, HID_CH><<<cdiv(waves, BLK / 32), BLK, 0, stream>>>(x, W1, hbuf, N);
  }

  // 3) agg1 = self-loop + bias, then edge scatter, then ReLU
  k_agg_init<HID_CH><<<cdiv((long long)N * HID_CH, BLK), BLK, 0, stream>>>(hbuf, dinv, b1, agg1, N);
  k_edge_scatter<HID_CH><<<cdiv((long long)E * (HID_CH / 4), BLK), BLK, 0, stream>>>(
      src, dst, ew, dinv, hbuf, agg1, E);
  k_relu<<<cdiv((long long)N * HID_CH, BLK), BLK, 0, stream>>>(agg1, N * HID_CH);

  // 4) h2 = relu(agg1) @ W2   (reuse hbuf)
  {
    const int waves = (N / 16) * (OUT_CH / 16);
    k_gemm_wmma<HID_CH, OUT_CH><<<cdiv(waves, BLK / 32), BLK, 0, stream>>>(agg1, W2, hbuf, N);
  }

  // 5) out = self-loop + bias, then edge scatter
  k_agg_init<OUT_CH><<<cdiv((long long)N * OUT_CH, BLK), BLK, 0, stream>>>(hbuf, dinv, b2, out, N);
  k_edge_scatter<OUT_CH><<<cdiv((long long)E * (OUT_CH / 4), BLK), BLK, 0, stream>>>(
      src, dst, ew, dinv, hbuf, out, E);
}